// JointNLLLoss_74560632258982
// MI455X (gfx1250) — compile-verified
//
#include <hip/hip_runtime.h>

// Problem constants (from reference)
#define KK 6
#define AA 50000
#define TT 60
#define NN (AA * TT)          // 3,000,000 (a,t) cells
#define NPAIR (NN / 2)        // 1,500,000 float4-sized pairs
#define PLANE4 (NN / 2)       // float4 elements per K-plane of loc/scale (N*2 floats / 4)
#define NBLK 1024
#define NTHR 256

typedef float v2f __attribute__((ext_vector_type(2)));
typedef float v8f __attribute__((ext_vector_type(8)));

// ---------------------------------------------------------------------------
// Stage 1: streaming NLL accumulation. Memory-bound: 315 MB of fp32 streamed
// once via b128 loads. Writes [7][NBLK] partials (k=0..5 scene sums, [6]=count).
// ---------------------------------------------------------------------------
__global__ __launch_bounds__(NTHR) void nll_stage1(
    const float4* __restrict__ loc4,     // [K * PLANE4]
    const float4* __restrict__ scl4,     // [K * PLANE4]
    const float4* __restrict__ tgt4,     // [NPAIR]
    const unsigned char* __restrict__ msk, // [NN] bool bytes
    float* __restrict__ partial)         // [7 * NBLK]
{
    float acc[7] = {0.f, 0.f, 0.f, 0.f, 0.f, 0.f, 0.f};

    const int stride = gridDim.x * blockDim.x;
    for (int p = blockIdx.x * blockDim.x + threadIdx.x; p < NPAIR; p += stride) {
        const float4 tg = tgt4[p];
        const unsigned char mb0 = msk[2 * p];
        const unsigned char mb1 = msk[2 * p + 1];
        const float m0 = mb0 ? 1.0f : 0.0f;
        const float m1 = mb1 ? 1.0f : 0.0f;
        acc[6] += m0 + m1;

#pragma unroll
        for (int k = 0; k < KK; ++k) {
            const size_t base = (size_t)k * PLANE4 + (size_t)p;
            const float4 l = loc4[base];
            const float4 s = scl4[base];
            // clip scales
            const float sx0 = fmaxf(s.x, 1e-6f);
            const float sy0 = fmaxf(s.y, 1e-6f);
            const float sx1 = fmaxf(s.z, 1e-6f);
            const float sy1 = fmaxf(s.w, 1e-6f);
            // log(2sx)+log(2sy) == log(4*sx*sy): one transcendental per (k,cell)
            const float n0 = __logf(4.0f * sx0 * sy0)
                           + fabsf(tg.x - l.x) / sx0
                           + fabsf(tg.y - l.y) / sy0;
            const float n1 = __logf(4.0f * sx1 * sy1)
                           + fabsf(tg.z - l.z) / sx1
                           + fabsf(tg.w - l.w) / sy1;
            acc[k] += m0 * n0 + m1 * n1;
        }
    }

    // wave32 shuffle reduction (deterministic order)
#pragma unroll
    for (int c = 0; c < 7; ++c) {
#pragma unroll
        for (int off = 16; off > 0; off >>= 1)
            acc[c] += __shfl_xor(acc[c], off, 32);
    }

    __shared__ float lds[NTHR / 32][7];
    const int wave = threadIdx.x >> 5;
    const int lane = threadIdx.x & 31;
    if (lane == 0) {
#pragma unroll
        for (int c = 0; c < 7; ++c) lds[wave][c] = acc[c];
    }
    __syncthreads();

    if (threadIdx.x == 0) {
#pragma unroll
        for (int c = 0; c < 7; ++c) {
            float s = 0.f;
#pragma unroll
            for (int w = 0; w < NTHR / 32; ++w) s += lds[w][c];
            partial[c * NBLK + blockIdx.x] = s;
        }
    }
}

// ---------------------------------------------------------------------------
// Stage 2: reduce [7][NBLK] partials. Last 64->1 step per channel done with
// v_wmma_f32_16x16x4_f32: A = all-ones => D[m][n] = sum_k B[k][n] for every m,
// so summing one D row over its 16 columns equals the sum of ALL 64 B slots,
// independent of the exact B VGPR slot mapping.
// ---------------------------------------------------------------------------
__global__ __launch_bounds__(NTHR) void nll_stage2(
    const float* __restrict__ partial,   // [7 * NBLK]
    float* __restrict__ out)             // [1]
{
    __shared__ float lds[7][NTHR];
    __shared__ float lds2[7][64];

    const int t = threadIdx.x;

    float acc[7];
#pragma unroll
    for (int c = 0; c < 7; ++c) acc[c] = 0.f;
    for (int j = t; j < NBLK; j += NTHR) {
#pragma unroll
        for (int c = 0; c < 7; ++c) acc[c] += partial[c * NBLK + j];
    }
#pragma unroll
    for (int c = 0; c < 7; ++c) lds[c][t] = acc[c];
    __syncthreads();

    if (t < 64) {
#pragma unroll
        for (int c = 0; c < 7; ++c)
            lds2[c][t] = lds[c][t] + lds[c][t + 64] + lds[c][t + 128] + lds[c][t + 192];
    }
    __syncthreads();

    // Uniform control flow from here: every wave runs the WMMA with EXEC all-1s.
    const int lane = t & 31;
    const v2f ones = {1.0f, 1.0f};
    float tot[7];
#pragma unroll
    for (int c = 0; c < 7; ++c) {
        v2f b;
        b.x = lds2[c][lane];        // 64 values -> 2 B-operand VGPRs, any slot order
        b.y = lds2[c][lane + 32];
        v8f cz = {};
        // D = ones(16x4) x B(4x16) + 0 : every D element is a column sum of B
        v8f d = __builtin_amdgcn_wmma_f32_16x16x4_f32(
            /*neg_a=*/false, ones, /*neg_b=*/false, b,
            /*c_mod=*/(short)0, cz, /*reuse_a=*/false, /*reuse_b=*/false);
        // Row 0 of D lives in d[0], lanes 0..15 (lanes 16..31 hold the identical
        // row 8). Keep lanes 0..15, sum across the wave -> sum of all 64 B slots.
        float r = (lane < 16) ? d[0] : 0.0f;
#pragma unroll
        for (int off = 16; off > 0; off >>= 1)
            r += __shfl_xor(r, off, 32);
        tot[c] = r;
    }

    if (t == 0) {
        float winner = tot[0];
#pragma unroll
        for (int k = 1; k < KK; ++k) winner = fminf(winner, tot[k]);
        const float valid = fmaxf(tot[6], 1.0f);
        out[0] = winner / valid;
    }
}

// ---------------------------------------------------------------------------
extern "C" void kernel_launch(void* const* d_in, const int* in_sizes, int n_in,
                              void* d_out, int out_size, void* d_ws, size_t ws_size,
                              hipStream_t stream) {
    const float4* loc4 = (const float4*)d_in[0];        // pred_loc   [K,A,T,2] f32
    const float4* scl4 = (const float4*)d_in[1];        // pred_scale [K,A,T,2] f32
    const float4* tgt4 = (const float4*)d_in[2];        // target     [A,T,2]   f32
    // d_in[3] = agent_batch (int64) is mathematically unused (full-sum identity)
    const unsigned char* msk = (const unsigned char*)d_in[4]; // mask [A,T] bool

    float* partial = (float*)d_ws;                      // 7 * NBLK * 4 = 28 KB

    nll_stage1<<<NBLK, NTHR, 0, stream>>>(loc4, scl4, tgt4, msk, partial);
    nll_stage2<<<1, NTHR, 0, stream>>>(partial, (float*)d_out);
}